// SpatialGCN_11579231830106
// MI455X (gfx1250) — compile-verified
//
#include <hip/hip_runtime.h>
#include <hip/hip_bf16.h>

typedef __attribute__((ext_vector_type(2))) float v2f;
typedef __attribute__((ext_vector_type(8))) float v8f;

#define N_NODES 100000
#define N_EDGES 640000
#define IN_C    128
#define HID_C   256
#define OUT_C   128
#define N_GRAPHS 64

// ---------------------------------------------------------------------------
// Utility: zero a float span (grid-stride)
// ---------------------------------------------------------------------------
__global__ void zero_f32(float* __restrict__ p, long n) {
    long i = (long)blockIdx.x * blockDim.x + threadIdx.x;
    long stride = (long)gridDim.x * blockDim.x;
    for (; i < n; i += stride) p[i] = 0.0f;
}

// ---------------------------------------------------------------------------
// Degree: deg[dst[e]] += 1  (float atomics; deg buffer pre-zeroed)
// ---------------------------------------------------------------------------
__global__ void degree_kernel(const int* __restrict__ dst, float* __restrict__ deg, int E) {
    int e = blockIdx.x * blockDim.x + threadIdx.x;
    if (e < E) atomicAdd(&deg[dst[e]], 1.0f);
}

// isd[i] = rsqrt(deg[i] + 1)   (in-place over the deg buffer)
__global__ void isd_kernel(float* __restrict__ deg, int n) {
    int i = blockIdx.x * blockDim.x + threadIdx.x;
    if (i < n) deg[i] = rsqrtf(deg[i] + 1.0f);
}

// ---------------------------------------------------------------------------
// FP32 WMMA GEMM:  C[M,N] = A[M,K] * B[K,N]
// One wave32 computes one 16x16 output tile via V_WMMA_F32_16X16X4_F32.
//   A 16x4 f32 frag (2 VGPR): lane m = lane&15; lanes 0-15 hold K=0,1;
//                             lanes 16-31 hold K=2,3  (khalf = (lane>>4)*2)
//   B 4x16 f32 frag (2 VGPR): lane n = lane&15; same K-half split.
//   C/D 16x16 f32 (8 VGPR):   vgpr r -> M=r (lanes 0-15), M=r+8 (lanes 16-31)
// Wave-uniform bounds guard keeps EXEC all-1s for every executed WMMA.
// ---------------------------------------------------------------------------
__global__ __launch_bounds__(256) void gemm_wmma_f32(
        const float* __restrict__ A, const float* __restrict__ B,
        float* __restrict__ C, int M, int N, int K) {
    const int lane = threadIdx.x & 31;
    const int waveInBlk = threadIdx.x >> 5;
    const long tile = (long)blockIdx.x * (blockDim.x >> 5) + waveInBlk;
    const int tilesN = N >> 4;
    const long totalTiles = (long)(M >> 4) * tilesN;
    if (tile >= totalTiles) return;              // wave-uniform: EXEC stays full

    const int tm = (int)(tile / tilesN);
    const int tn = (int)(tile % tilesN);
    const int mrow = tm * 16 + (lane & 15);      // A row this lane loads
    const int ncol = tn * 16 + (lane & 15);      // B/C/D column this lane owns
    const int khalf = (lane >> 4) * 2;           // which K-pair this half-wave holds

    v8f acc = {};
    const float* arow = A + (long)mrow * K;
    for (int k = 0; k < K; k += 4) {
        v2f a = *(const v2f*)(arow + k + khalf);                 // K contiguous pair
        v2f b;
        b.x = B[(long)(k + khalf)     * N + ncol];
        b.y = B[(long)(k + khalf + 1) * N + ncol];
        acc = __builtin_amdgcn_wmma_f32_16x16x4_f32(
                  /*neg_a=*/false, a, /*neg_b=*/false, b,
                  /*c_mod=*/(short)0, acc, /*reuse_a=*/false, /*reuse_b=*/false);
    }

    const int rowBase = tm * 16 + (lane >> 4) * 8;
    #pragma unroll
    for (int r = 0; r < 8; ++r)
        C[(long)(rowBase + r) * N + ncol] = acc[r];
}

// ---------------------------------------------------------------------------
// Edge scatter: agg[dst, c] += h[src, c] * isd[src] * isd[dst]
// One block per edge, one thread per channel (coalesced gather + atomic scatter)
// ---------------------------------------------------------------------------
__global__ void scatter_edges(const float* __restrict__ h,
                              const int* __restrict__ src, const int* __restrict__ dst,
                              const float* __restrict__ isd,
                              float* __restrict__ agg, int C) {
    const int e = blockIdx.x;
    const int c = threadIdx.x;
    const int s = src[e];
    const int d = dst[e];
    const float coef = isd[s] * isd[d];
    atomicAdd(&agg[(long)d * C + c], h[(long)s * C + c] * coef);
}

// ---------------------------------------------------------------------------
// Layer finalize (in place): agg = relu(agg + h*isd^2 + b)
// ---------------------------------------------------------------------------
__global__ void finalize_relu(float* __restrict__ agg, const float* __restrict__ h,
                              const float* __restrict__ isd, const float* __restrict__ b,
                              long total, int C) {
    long i = (long)blockIdx.x * blockDim.x + threadIdx.x;
    if (i >= total) return;
    const long node = i / C;
    const int  c    = (int)(i - node * C);
    const float s = isd[node];
    float v = agg[i] + h[i] * (s * s) + b[c];
    agg[i] = fmaxf(v, 0.0f);
}

// ---------------------------------------------------------------------------
// Final layer: out = relu(out + h*isd^2 + b) (out pre-holds edge aggregation),
// plus atomic graph pooling (sum per graph + node counts).
// ---------------------------------------------------------------------------
__global__ void finalize2_pool(float* __restrict__ out, const float* __restrict__ h,
                               const float* __restrict__ isd, const float* __restrict__ b,
                               const int* __restrict__ batch,
                               float* __restrict__ poolSum, float* __restrict__ poolCnt,
                               long total, int C) {
    long i = (long)blockIdx.x * blockDim.x + threadIdx.x;
    if (i >= total) return;
    const long node = i / C;
    const int  c    = (int)(i - node * C);
    const float s = isd[node];
    float v = fmaxf(out[i] + h[i] * (s * s) + b[c], 0.0f);
    out[i] = v;
    const int g = batch[node];
    atomicAdd(&poolSum[(long)g * C + c], v);
    if (c == 0) atomicAdd(&poolCnt[g], 1.0f);
}

// graph_embed[g,c] = poolSum / max(cnt, 1)
__global__ void pool_div(const float* __restrict__ poolSum, const float* __restrict__ poolCnt,
                         float* __restrict__ embed, int C) {
    int i = blockIdx.x * blockDim.x + threadIdx.x;
    if (i >= N_GRAPHS * C) return;
    int g = i / C;
    embed[i] = poolSum[i] / fmaxf(poolCnt[g], 1.0f);
}

// ---------------------------------------------------------------------------
extern "C" void kernel_launch(void* const* d_in, const int* in_sizes, int n_in,
                              void* d_out, int out_size, void* d_ws, size_t ws_size,
                              hipStream_t stream) {
    const float* x   = (const float*)d_in[0];
    const int*   ei  = (const int*)  d_in[1];
    const int*   bat = (const int*)  d_in[2];
    const float* W1  = (const float*)d_in[3];
    const float* b1  = (const float*)d_in[4];
    const float* W2  = (const float*)d_in[5];
    const float* b2  = (const float*)d_in[6];
    float* out = (float*)d_out;

    const int*  src = ei;
    const int*  dst = ei + N_EDGES;

    // Workspace layout (floats)
    float* ws   = (float*)d_ws;
    float* isd  = ws;                                   // N_NODES (deg -> isd)
    float* h1   = isd + 102400;                         // N_NODES*HID_C
    float* agg1 = h1  + (long)N_NODES * HID_C;          // N_NODES*HID_C
    float* h2   = agg1 + (long)N_NODES * HID_C;         // N_NODES*OUT_C
    float* poolSum = h2 + (long)N_NODES * OUT_C;        // N_GRAPHS*OUT_C
    float* poolCnt = poolSum + N_GRAPHS * OUT_C;        // N_GRAPHS

    float* outH = out;                                  // [N_NODES, OUT_C]
    float* outE = out + (long)N_NODES * OUT_C;          // [N_GRAPHS, OUT_C]

    auto cdiv = [](long a, long b) { return (int)((a + b - 1) / b); };

    // --- zero accumulators (required every call) ---
    zero_f32<<<512,  256, 0, stream>>>(isd, N_NODES);
    zero_f32<<<2048, 256, 0, stream>>>(agg1, (long)N_NODES * HID_C);
    zero_f32<<<2048, 256, 0, stream>>>(outH, (long)N_NODES * OUT_C);
    zero_f32<<<34,   256, 0, stream>>>(poolSum, N_GRAPHS * OUT_C + N_GRAPHS);

    // --- degrees + normalization ---
    degree_kernel<<<cdiv(N_EDGES, 256), 256, 0, stream>>>(dst, isd, N_EDGES);
    isd_kernel<<<cdiv(N_NODES, 256), 256, 0, stream>>>(isd, N_NODES);

    // --- layer 1: h1 = x @ W1 (fp32 WMMA), scatter, finalize+relu ---
    {
        long tiles = ((long)N_NODES / 16) * (HID_C / 16);
        gemm_wmma_f32<<<cdiv(tiles, 8), 256, 0, stream>>>(x, W1, h1, N_NODES, HID_C, IN_C);
    }
    scatter_edges<<<N_EDGES, HID_C, 0, stream>>>(h1, src, dst, isd, agg1, HID_C);
    finalize_relu<<<cdiv((long)N_NODES * HID_C, 256), 256, 0, stream>>>(
        agg1, h1, isd, b1, (long)N_NODES * HID_C, HID_C);

    // --- layer 2: h2 = agg1 @ W2, scatter into d_out, finalize+relu+pool ---
    {
        long tiles = ((long)N_NODES / 16) * (OUT_C / 16);
        gemm_wmma_f32<<<cdiv(tiles, 8), 256, 0, stream>>>(agg1, W2, h2, N_NODES, OUT_C, HID_C);
    }
    scatter_edges<<<N_EDGES, OUT_C, 0, stream>>>(h2, src, dst, isd, outH, OUT_C);
    finalize2_pool<<<cdiv((long)N_NODES * OUT_C, 256), 256, 0, stream>>>(
        outH, h2, isd, b2, bat, poolSum, poolCnt, (long)N_NODES * OUT_C, OUT_C);

    // --- mean pool ---
    pool_div<<<cdiv(N_GRAPHS * OUT_C, 256), 256, 0, stream>>>(poolSum, poolCnt, outE, OUT_C);
}